// MultiFidelityTM_34299608826384
// MI455X (gfx1250) — compile-verified
//
#include <hip/hip_runtime.h>
#include <math.h>

typedef __attribute__((ext_vector_type(2))) float v2f;
typedef __attribute__((ext_vector_type(8))) float v8f;

#define RR        2
#define N_PER_FID 4096
#define N_REP     128
#define MDIM      30
#define KDIM      60
#define NUG_MULT  4.0f
#define JITTER    1e-6f
#define NB        16            // Cholesky panel width (one WMMA tile)
#define NPANEL    (N_REP / NB)  // 8

// LDS layout (floats). Strides padded: 65 -> bank (m+k)%64 conflict-free on
// column walks; 129 -> bank (i+k)%64 conflict-free for Cholesky column ops.
#define XS_STRIDE 65
#define GS_STRIDE 129
#define XS_OFF    0
#define GS_OFF    (N_REP * XS_STRIDE)            // 8320
#define SC_OFF    (GS_OFF + N_REP * GS_STRIDE)   // 24832
#define DIAG_OFF  (SC_OFF + 64)
#define YV_OFF    (DIAG_OFF + 128)
#define RED_OFF   (YV_OFF + 128)
#define SMEM_FLOATS (RED_OFF + 256)              // 25408 floats = 101632 B

__global__ __launch_bounds__(256)
void mftm_kernel(const float* __restrict__ response,
                 const float* __restrict__ x_now,
                 const float* __restrict__ x_preb,
                 const float* __restrict__ scales,
                 const float* __restrict__ nugget_params,
                 const float* __restrict__ sigma_params,
                 const float* __restrict__ theta_q,
                 const float* __restrict__ lengthscale,
                 const int*   __restrict__ rptr,
                 float* __restrict__ out)
{
    extern __shared__ float smem[];
    float* Xs   = smem + XS_OFF;
    float* Gs   = smem + GS_OFF;
    float* scv  = smem + SC_OFF;
    float* diag = smem + DIAG_OFF;
    float* yv   = smem + YV_OFF;
    float* red  = smem + RED_OFF;

    const int t = threadIdx.x;
    const int b = blockIdx.x;
    const int r = rptr[0];
    const int start = r * N_PER_FID;

    // ---- column scaling vector: scal_now (k=1..30), scal_preb (k=1..30) ----
    if (t < KDIM) {
        float kf = (float)((t % MDIM) + 1);
        float a0, a1;
        if (t < MDIM) { a0 = theta_q[r];              a1 = theta_q[r + RR]; }
        else          { a0 = theta_q[r + 2 * RR - 1]; a1 = theta_q[r + 3 * RR - 2]; }
        scv[t] = __expf(a0 - 0.5f * __expf(a1) * kf);
    }
    __syncthreads();

    // ---- load + scale X (128 x 60) into LDS (single HBM pass over x) ----
    const float* xn = x_now  + (size_t)b * (N_REP * MDIM);
    const float* xp = x_preb + (size_t)b * (N_REP * MDIM);
    for (int f = t; f < N_REP * MDIM; f += 256) {
        int i = f / MDIM;
        int m = f - i * MDIM;
        Xs[i * XS_STRIDE + m]        = xn[f] * scv[m];
        Xs[i * XS_STRIDE + MDIM + m] = xp[f] * scv[MDIM + m];
    }
    __syncthreads();

    // ---- Gram S = X X^T via V_WMMA_F32_16X16X4_F32 ----
    // wave w owns tile-row w; 8 tiles of 16x16, K=60 -> 15 chained WMMAs.
    const int w     = t >> 5;          // wave id = tile row
    const int l     = t & 31;          // lane
    const int m16   = l & 15;          // matrix row/col within tile
    const int khalf = (l >> 4) << 1;   // K sub-offset per ISA A/B layout (0 or 2)
    const int dvrow = (l >> 4) << 3;   // D-layout VGPR row base offset (0 or 8)

    {
        const float* arow = &Xs[(w * 16 + m16) * XS_STRIDE];
        for (int tj = 0; tj < 8; ++tj) {
            const float* brow = &Xs[(tj * 16 + m16) * XS_STRIDE];
            v8f acc = {};
#pragma unroll
            for (int kk = 0; kk < 15; ++kk) {
                int kb = kk * 4 + khalf;
                v2f av; av.x = arow[kb]; av.y = arow[kb + 1];
                v2f bv; bv.x = brow[kb]; bv.y = brow[kb + 1];
                acc = __builtin_amdgcn_wmma_f32_16x16x4_f32(
                    false, av, false, bv, (short)0, acc, false, false);
            }
            int rbase = w * 16 + dvrow;
            int c     = tj * 16 + m16;
#pragma unroll
            for (int v = 0; v < 8; ++v)
                Gs[(rbase + v) * GS_STRIDE + c] = acc[v];
        }
    }
    __syncthreads();

    // ---- per-batch scalars ----
    float log_sc = __logf(scales[start + b]);
    float nug = __expf(nugget_params[r] + nugget_params[r + RR] * log_sc);
    nug = fmaxf(nug - 1e-5f, 0.0f) + 1e-5f;
    float sig  = __expf(sigma_params[r] + sigma_params[r + RR] * log_sc);
    float sig2 = sig * sig;
    float ls   = __expf(lengthscale[r]) * sqrtf(3.0f);  // sqrt(2*SMOOTH)
    float inv_ls2 = 1.0f / (ls * ls);
    float inv_nug = 1.0f / nug;

    // ---- Matern-1.5 from Gram diagonal, assemble G in place ----
    if (t < N_REP) diag[t] = Gs[t * GS_STRIDE + t];
    __syncthreads();
    for (int e = t; e < N_REP * N_REP; e += 256) {
        int i = e >> 7;
        int j = e & 127;
        float s   = Gs[i * GS_STRIDE + j];
        float d2  = fmaxf((diag[i] + diag[j] - 2.0f * s) * inv_ls2, 0.0f);
        float s3d = sqrtf(3.0f * d2);
        float mat = (1.0f + s3d) * __expf(-s3d);
        float g   = (s + sig2 * mat) * inv_nug;
        if (i == j) g += 1.0f + JITTER;
        Gs[i * GS_STRIDE + j] = g;
    }
    __syncthreads();

    // ---- blocked Cholesky: 8 panels of 16; trailing SYRK on the WMMA pipe ----
    for (int p = 0; p < NPANEL; ++p) {
        const int base = p * NB;

        // (1) 16x16 diagonal-block micro-Cholesky: wave 0 only.
        //     Relies on wave32 lockstep + in-order DS ops (no barriers needed).
        if (w == 0) {
            for (int k = 0; k < NB; ++k) {
                float dkk = Gs[(base + k) * GS_STRIDE + base + k];
                float lkk = sqrtf(dkk);
                if (l == k) Gs[(base + k) * GS_STRIDE + base + k] = lkk;
                if (l > k && l < NB) {
                    float lik = Gs[(base + l) * GS_STRIDE + base + k] / lkk;
                    Gs[(base + l) * GS_STRIDE + base + k] = lik;
                    for (int j = k + 1; j <= l; ++j)
                        Gs[(base + l) * GS_STRIDE + base + j] -=
                            lik * Gs[(base + j) * GS_STRIDE + base + k];
                }
            }
        }
        __syncthreads();

        // (2) panel triangular solve: one thread per trailing row,
        //     fully-unrolled 16-wide forward substitution in registers.
        const int nrows = N_REP - base - NB;
        if (t < nrows) {
            const int i = base + NB + t;
            float x[NB];
#pragma unroll
            for (int j = 0; j < NB; ++j) {
                float v = Gs[i * GS_STRIDE + base + j];
#pragma unroll
                for (int q = 0; q < NB; ++q)
                    if (q < j) v -= x[q] * Gs[(base + j) * GS_STRIDE + base + q];
                x[j] = v / Gs[(base + j) * GS_STRIDE + base + j];
            }
#pragma unroll
            for (int j = 0; j < NB; ++j)
                Gs[i * GS_STRIDE + base + j] = x[j];
        }
        __syncthreads();

        // (3) trailing update G -= P * P^T : rank-16 SYRK via WMMA.
        //     A-fragment negated at load (f32 WMMA NEG only covers C).
        const int ntr = NPANEL - 1 - p;
        for (int idx = w; idx < ntr * ntr; idx += 8) {
            const int ti = p + 1 + idx / ntr;
            const int tj = p + 1 + idx % ntr;
            const int rbase = ti * 16 + dvrow;
            const int c     = tj * 16 + m16;
            v8f acc;
#pragma unroll
            for (int v = 0; v < 8; ++v)
                acc[v] = Gs[(rbase + v) * GS_STRIDE + c];
            const float* arow = &Gs[(ti * 16 + m16) * GS_STRIDE + base];
            const float* brow = &Gs[(tj * 16 + m16) * GS_STRIDE + base];
#pragma unroll
            for (int kk = 0; kk < 4; ++kk) {
                int kb = kk * 4 + khalf;
                v2f av; av.x = -arow[kb]; av.y = -arow[kb + 1];
                v2f bv; bv.x =  brow[kb]; bv.y =  brow[kb + 1];
                acc = __builtin_amdgcn_wmma_f32_16x16x4_f32(
                    false, av, false, bv, (short)0, acc, false, false);
            }
#pragma unroll
            for (int v = 0; v < 8; ++v)
                Gs[(rbase + v) * GS_STRIDE + c] = acc[v];
        }
        __syncthreads();
    }

    // ---- blocked forward solve L y = resp_b ----
    if (t < N_REP) yv[t] = response[(size_t)t * (RR * N_PER_FID) + start + b];
    __syncthreads();
    for (int p = 0; p < NPANEL; ++p) {
        const int base = p * NB;
        if (t == 0) {
#pragma unroll
            for (int j = 0; j < NB; ++j) {
                float v = yv[base + j];
#pragma unroll
                for (int q = 0; q < NB; ++q)
                    if (q < j) v -= Gs[(base + j) * GS_STRIDE + base + q] * yv[base + q];
                yv[base + j] = v / Gs[(base + j) * GS_STRIDE + base + j];
            }
        }
        __syncthreads();
        const int nrows = N_REP - base - NB;
        if (t < nrows) {
            const int i = base + NB + t;
            float acc = yv[i];
#pragma unroll
            for (int j = 0; j < NB; ++j)
                acc -= Gs[i * GS_STRIDE + base + j] * yv[base + j];
            yv[i] = acc;
        }
        __syncthreads();
    }

    // ---- reductions: sum(y^2), sum(log diag L) ----
    red[t] = (t < N_REP) ? (yv[t] * yv[t]) : 0.0f;
    __syncthreads();
    for (int s = 128; s > 0; s >>= 1) {
        if (t < s) red[t] += red[t + s];
        __syncthreads();
    }
    float sumy2 = red[0];
    __syncthreads();
    red[t] = (t < N_REP) ? __logf(Gs[t * GS_STRIDE + t]) : 0.0f;
    __syncthreads();
    for (int s = 128; s > 0; s >>= 1) {
        if (t < s) red[t] += red[t + s];
        __syncthreads();
    }

    if (t == 0) {
        float logdet     = red[0];
        float nug_sd     = nug * NUG_MULT;
        float alpha      = (nug * nug) / (nug_sd * nug_sd) + 2.0f;  // 2.0625
        float beta       = nug * (alpha - 1.0f);
        float alpha_post = alpha + 0.5f * (float)N_REP;
        float beta_post  = beta + 0.5f * sumy2;
        out[b] = -logdet + alpha * __logf(beta) - alpha_post * __logf(beta_post)
                 + lgammaf(alpha_post) - lgammaf(alpha);
    }
}

extern "C" void kernel_launch(void* const* d_in, const int* in_sizes, int n_in,
                              void* d_out, int out_size, void* d_ws, size_t ws_size,
                              hipStream_t stream) {
    const float* response      = (const float*)d_in[0];
    const float* x_now         = (const float*)d_in[1];
    const float* x_preb        = (const float*)d_in[2];
    const float* scales        = (const float*)d_in[3];
    const float* nugget_params = (const float*)d_in[4];
    const float* sigma_params  = (const float*)d_in[5];
    const float* theta_q       = (const float*)d_in[6];
    const float* lengthscale   = (const float*)d_in[7];
    const int*   rptr          = (const int*)d_in[8];
    float* out = (float*)d_out;

    (void)in_sizes; (void)n_in; (void)out_size; (void)d_ws; (void)ws_size;

    size_t shmem = (size_t)SMEM_FLOATS * sizeof(float);   // ~101.6 KB of WGP LDS
    mftm_kernel<<<N_PER_FID, 256, shmem, stream>>>(
        response, x_now, x_preb, scales, nugget_params, sigma_params,
        theta_q, lengthscale, rptr, out);
}